// BGATTNET_Loss_23502061044109
// MI455X (gfx1250) — compile-verified
//
#include <hip/hip_runtime.h>
#include <hip/hip_bf16.h>
#include <math.h>

typedef __attribute__((ext_vector_type(2))) float v2f;
typedef __attribute__((ext_vector_type(8))) float v8f;

#define B_ 8
#define N_ 2048
#define D_ 512
#define CHUNKS 16
#define ROWS_PER_CHUNK (N_ / CHUNKS)            // 128
#define WAVES 8
#define ROWS_PER_WAVE (ROWS_PER_CHUNK / WAVES)  // 16
#define COE_ 0.01f
#define EPS_ 1e-12f

// Workspace layout (in floats):
//   [0]                          partial_s   [B_*CHUNKS][D_]
//   [WS_PARTIAL_DIAG]            partial_diag[B_*CHUNKS]
//   [WS_S]                       s           [16][D_]   (rows 8..15 zero)
//   [WS_DIAG]                    diagSum     [B_]
#define WS_PARTIAL_DIAG (B_ * CHUNKS * D_)
#define WS_S (WS_PARTIAL_DIAG + B_ * CHUNKS)
#define WS_DIAG (WS_S + 16 * D_)

__device__ __forceinline__ float sq4(float4 a) {
  return a.x * a.x + a.y * a.y + a.z * a.z + a.w * a.w;
}

// ---------------------------------------------------------------------------
// Kernel 1: stream H once. Per row: ||h||^2 (wave reduce), scale by
// 1/max(||h||,eps), accumulate normalized-row column sums in VGPRs, then a
// fixed-order LDS cross-wave reduction -> per-block partials in ws.
// Memory-bound: 32 MB total read, float4 coalesced.
// ---------------------------------------------------------------------------
__global__ __launch_bounds__(256) void rownorm_partial(
    const float* __restrict__ H, float* __restrict__ ws) {
  const int blk = blockIdx.x;            // 0 .. B_*CHUNKS-1
  const int b = blk / CHUNKS;
  const int chunk = blk % CHUNKS;
  const int wave = threadIdx.x >> 5;
  const int lane = threadIdx.x & 31;

  float4 acc0 = {0.f, 0.f, 0.f, 0.f};
  float4 acc1 = acc0, acc2 = acc0, acc3 = acc0;
  float diag_acc = 0.f;

  const int row0 = chunk * ROWS_PER_CHUNK + wave * ROWS_PER_WAVE;
#pragma unroll 4
  for (int i = 0; i < ROWS_PER_WAVE; ++i) {
    const int r = row0 + i;
    const float4* p =
        reinterpret_cast<const float4*>(H + ((size_t)b * N_ + r) * D_);
    // lane l covers elements d = 4*l + 128*q, q = 0..3 (fully coalesced b128)
    float4 x0 = p[lane + 0];
    float4 x1 = p[lane + 32];
    float4 x2 = p[lane + 64];
    float4 x3 = p[lane + 96];

    float ss = sq4(x0) + sq4(x1) + sq4(x2) + sq4(x3);
#pragma unroll
    for (int off = 16; off >= 1; off >>= 1) ss += __shfl_xor(ss, off, 32);

    const float norm = sqrtf(ss);
    const float rinv = 1.0f / fmaxf(norm, EPS_);
    diag_acc += ss * rinv * rinv;  // == 1 unless degenerate (F.normalize eps)

    acc0.x += x0.x * rinv; acc0.y += x0.y * rinv; acc0.z += x0.z * rinv; acc0.w += x0.w * rinv;
    acc1.x += x1.x * rinv; acc1.y += x1.y * rinv; acc1.z += x1.z * rinv; acc1.w += x1.w * rinv;
    acc2.x += x2.x * rinv; acc2.y += x2.y * rinv; acc2.z += x2.z * rinv; acc2.w += x2.w * rinv;
    acc3.x += x3.x * rinv; acc3.y += x3.y * rinv; acc3.z += x3.z * rinv; acc3.w += x3.w * rinv;
  }

  __shared__ float red[WAVES][D_];
  __shared__ float wdiag[WAVES];
  *(float4*)&red[wave][lane * 4 + 0]   = acc0;
  *(float4*)&red[wave][lane * 4 + 128] = acc1;
  *(float4*)&red[wave][lane * 4 + 256] = acc2;
  *(float4*)&red[wave][lane * 4 + 384] = acc3;
  if (lane == 0) wdiag[wave] = diag_acc;
  __syncthreads();

  for (int d = threadIdx.x; d < D_; d += 256) {
    float sum = 0.f;
#pragma unroll
    for (int w = 0; w < WAVES; ++w) sum += red[w][d];  // fixed order
    ws[(size_t)blk * D_ + d] = sum;
  }
  if (threadIdx.x == 0) {
    float s = 0.f;
#pragma unroll
    for (int w = 0; w < WAVES; ++w) s += wdiag[w];
    ws[WS_PARTIAL_DIAG + blk] = s;
  }
}

// ---------------------------------------------------------------------------
// Kernel 2: fixed-order reduction of chunk partials -> s[16][D_] (zero-padded
// to 16 rows for the WMMA tile) and diagSum[B_].
// ---------------------------------------------------------------------------
__global__ __launch_bounds__(512) void reduce_partials(float* __restrict__ ws) {
  const int b = blockIdx.x;   // 0..15
  const int d = threadIdx.x;  // 0..511
  float v = 0.f;
  if (b < B_) {
#pragma unroll
    for (int c = 0; c < CHUNKS; ++c)
      v += ws[(size_t)(b * CHUNKS + c) * D_ + d];
  }
  ws[WS_S + b * D_ + d] = v;
  if (threadIdx.x == 0 && b < B_) {
    float s = 0.f;
#pragma unroll
    for (int c = 0; c < CHUNKS; ++c) s += ws[WS_PARTIAL_DIAG + b * CHUNKS + c];
    ws[WS_DIAG + b] = s;
  }
}

// ---------------------------------------------------------------------------
// Kernel 3 (one wave32): G = s . s^T via 128x V_WMMA_F32_16X16X4_F32.
// For a 16x4 f32 A-tile: lane L<16 holds (row L, K+0/K+1); lane L>=16 holds
// (row L-16, K+2/K+3). The 4x16 B-tile = A^T has the mirrored lane layout,
// so per-lane A and B register contents coincide. Diagonal of G gives
// ||s_b||^2. Then reg term + cross-entropy -> single scalar output.
// ---------------------------------------------------------------------------
__global__ __launch_bounds__(32) void finalize(
    const float* __restrict__ outputs, const long long* __restrict__ labels,
    const float* __restrict__ ws, float* __restrict__ out) {
  const int lane = threadIdx.x;
  const float* S = ws + WS_S;
  __shared__ float G[16][16];

#if __has_builtin(__builtin_amdgcn_wmma_f32_16x16x4_f32)
  v8f c = {0.f, 0.f, 0.f, 0.f, 0.f, 0.f, 0.f, 0.f};
  const int row = lane & 15;
  const int ksel = (lane < 16) ? 0 : 2;  // VALU select, no divergence
  for (int k0 = 0; k0 < D_; k0 += 4) {
    v2f a;
    a.x = S[row * D_ + k0 + ksel];
    a.y = S[row * D_ + k0 + ksel + 1];
    // (neg_a, A, neg_b, B, c_mod, C, reuse_a, reuse_b); B-tile == A^T shares
    // per-lane registers with A for this symmetric Gram.
    c = __builtin_amdgcn_wmma_f32_16x16x4_f32(false, a, false, a, (short)0, c,
                                              false, false);
  }
  // C/D layout: VGPR r -> M=r (lanes 0-15) / M=r+8 (lanes 16-31), N = lane&15
#pragma unroll
  for (int r = 0; r < 8; ++r) {
    const int m = (lane < 16) ? r : (r + 8);
    G[m][lane & 15] = c[r];
  }
  __syncthreads();
#else
  // VALU fallback (host pass / missing builtin): only the diagonal is needed.
  if (lane < 16) {
    float g = 0.f;
    for (int d = 0; d < D_; ++d) {
      const float v = S[lane * D_ + d];
      g += v * v;
    }
    G[lane][lane] = g;
  }
  __syncthreads();
#endif

  float partial = 0.f;
  if (lane < B_) {
    const float n_pairs = (float)N_ * (float)(N_ - 1) * 0.5f;
    partial = (G[lane][lane] - ws[WS_DIAG + lane]) * (0.5f * COE_ / n_pairs);
  }
#pragma unroll
  for (int off = 16; off >= 1; off >>= 1)
    partial += __shfl_xor(partial, off, 32);

  if (lane == 0) {
    float ce = 0.f;
#pragma unroll
    for (int i = 0; i < B_; ++i) {
      const float o0 = outputs[2 * i + 0];
      const float o1 = outputs[2 * i + 1];
      const float m = fmaxf(o0, o1);
      const float lse = m + logf(expf(o0 - m) + expf(o1 - m));
      const int lbl = (int)labels[i];
      ce += lse - (lbl ? o1 : o0);
    }
    ce *= (1.0f / (float)B_);
    out[0] = ce + partial;
  }
}

// ---------------------------------------------------------------------------
extern "C" void kernel_launch(void* const* d_in, const int* in_sizes, int n_in,
                              void* d_out, int out_size, void* d_ws,
                              size_t ws_size, hipStream_t stream) {
  (void)in_sizes; (void)n_in; (void)out_size; (void)ws_size;
  const float* outputs = (const float*)d_in[0];
  const long long* labels = (const long long*)d_in[1];  // int64 labels
  const float* H = (const float*)d_in[2];
  float* ws = (float*)d_ws;   // needs ~296 KB
  float* out = (float*)d_out;

  rownorm_partial<<<B_ * CHUNKS, 256, 0, stream>>>(H, ws);
  reduce_partials<<<16, 512, 0, stream>>>(ws);
  finalize<<<1, 32, 0, stream>>>(outputs, labels, ws, out);
}